// ContrastiveLoss_70085276336524
// MI455X (gfx1250) — compile-verified
//
#include <hip/hip_runtime.h>
#include <hip/hip_bf16.h>
#include <math.h>

typedef __attribute__((ext_vector_type(2))) float v2f;
typedef __attribute__((ext_vector_type(8))) float v8f;

#define TEMP_   0.07f
#define EPS_    1e-8f
#define NB      64
#define NT      512
#define ND      512
#define NCHUNK  4              // blocks per batch in main pass
#define ROWS_PB 128            // 8 waves * 16 rows per block

// ---------------------------------------------------------------------------
// Kernel 1: per-batch argmax(mask) (first t with mask==1, else 0) + anchor
// rows normalized (audio->a_anchor, video->v_anchor). One block per batch.
// ---------------------------------------------------------------------------
__global__ void cl_anchors_kernel(const float* __restrict__ video,
                                  const float* __restrict__ audio,
                                  const int*   __restrict__ mask,
                                  int*   __restrict__ wsIdx,
                                  float* __restrict__ wsAnch /* [aA 64*512 | vA 64*512] */)
{
    __shared__ int   sMin[256];
    __shared__ float sRed[256];
    const int b   = blockIdx.x;
    const int tid = threadIdx.x;

    // first index where mask == 1 (argmax of a 0/1 int mask)
    int m0 = mask[b * NT + tid];
    int m1 = mask[b * NT + tid + 256];
    int cand = 0x7fffffff;
    if (m0 > 0)      cand = tid;
    else if (m1 > 0) cand = tid + 256;
    sMin[tid] = cand;
    __syncthreads();
    for (int s = 128; s > 0; s >>= 1) {
        if (tid < s) sMin[tid] = min(sMin[tid], sMin[tid + s]);
        __syncthreads();
    }
    int idx = sMin[0];
    if (idx == 0x7fffffff) idx = 0;
    if (tid == 0) wsIdx[b] = idx;

    const float* vrow = video + ((size_t)b * NT + idx) * ND;
    const float* arow = audio + ((size_t)b * NT + idx) * ND;
    float v0 = vrow[tid], v1 = vrow[tid + 256];
    float a0 = arow[tid], a1 = arow[tid + 256];

    __syncthreads();
    sRed[tid] = v0 * v0 + v1 * v1;
    __syncthreads();
    for (int s = 128; s > 0; s >>= 1) { if (tid < s) sRed[tid] += sRed[tid + s]; __syncthreads(); }
    float nv = sRed[0];
    __syncthreads();
    sRed[tid] = a0 * a0 + a1 * a1;
    __syncthreads();
    for (int s = 128; s > 0; s >>= 1) { if (tid < s) sRed[tid] += sRed[tid + s]; __syncthreads(); }
    float na = sRed[0];

    float invV = 1.0f / fmaxf(sqrtf(nv), EPS_);
    float invA = 1.0f / fmaxf(sqrtf(na), EPS_);
    float* aA = wsAnch;                 // normalized audio anchor
    float* vA = wsAnch + NB * ND;       // normalized video anchor
    aA[b * ND + tid]       = a0 * invA;
    aA[b * ND + tid + 256] = a1 * invA;
    vA[b * ND + tid]       = v0 * invV;
    vA[b * ND + tid + 256] = v1 * invV;
}

// ---------------------------------------------------------------------------
// Kernel 2: main pass. grid = (NCHUNK, NB), block = 256 (8 waves).
// Each wave owns a 16-row tile; K-loop over D in steps of 4 feeding
// V_WMMA_F32_16X16X4_F32: A = 16x4 tile of rows, B = 4x16 with columns
// {a_anchor, v_anchor, 0...}. Row norms via VALU + shfl_xor(16).
// ---------------------------------------------------------------------------
__global__ void cl_sims_kernel(const float* __restrict__ video,
                               const float* __restrict__ audio,
                               const int*   __restrict__ wsIdx,
                               const float* __restrict__ wsAnch,
                               float* __restrict__ wsPos,
                               float* __restrict__ wsNegA,
                               float* __restrict__ wsNegV)
{
    __shared__ float sAnch[2 * ND];     // [aA(512) | vA(512)] for this b
    __shared__ float sDotAv[8][16];     // video_row . a_anchor
    __shared__ float sDotVa[8][16];     // audio_row . v_anchor
    __shared__ float sNv[8][16];
    __shared__ float sNa[8][16];
    __shared__ float sEa[ROWS_PB];
    __shared__ float sEv[ROWS_PB];

    const int chunk = blockIdx.x;       // 0..NCHUNK-1
    const int b     = blockIdx.y;
    const int tid   = threadIdx.x;      // 0..255
    const int wv    = tid >> 5;         // wave id 0..7
    const int lane  = tid & 31;

    // stage both anchors into LDS
    sAnch[tid]       = wsAnch[b * ND + tid];
    sAnch[tid + 256] = wsAnch[b * ND + tid + 256];
    sAnch[tid + 512] = wsAnch[NB * ND + b * ND + tid];
    sAnch[tid + 768] = wsAnch[NB * ND + b * ND + tid + 256];
    __syncthreads();

    const int row0 = chunk * ROWS_PB + wv * 16;    // tile base (t index)
    const int m    = lane & 15;                    // A row / B column index
    const int koff = (lane >> 4) << 1;             // 0 or 2
    const float* vbase = video + ((size_t)b * NT + row0 + m) * ND + koff;
    const float* abase = audio + ((size_t)b * NT + row0 + m) * ND + koff;
    const float* banch = sAnch + ((m == 1) ? ND : 0) + koff;
    const bool   bOn   = (m < 2);

    v8f cAv = {}; v8f cAa = {};
    float nv = 0.0f, na = 0.0f;

    for (int k0 = 0; k0 < ND; k0 += 4) {
        v2f Av = *(const v2f*)(vbase + k0);
        v2f Aa = *(const v2f*)(abase + k0);
        v2f Bt = *(const v2f*)(banch + k0);
        v2f Bm;
        Bm[0] = bOn ? Bt[0] : 0.0f;
        Bm[1] = bOn ? Bt[1] : 0.0f;

        nv += Av[0] * Av[0] + Av[1] * Av[1];
        na += Aa[0] * Aa[0] + Aa[1] * Aa[1];

        // C += A(16x4 f32) x B(4x16 f32); col0 = .a_anchor, col1 = .v_anchor
        cAv = __builtin_amdgcn_wmma_f32_16x16x4_f32(false, Av, false, Bm,
                                                    (short)0, cAv, false, false);
        cAa = __builtin_amdgcn_wmma_f32_16x16x4_f32(false, Aa, false, Bm,
                                                    (short)0, cAa, false, false);
    }

    // combine K-halves of the norms (lane L and L^16 hold halves of row L&15)
    nv += __shfl_xor(nv, 16, 32);
    na += __shfl_xor(na, 16, 32);
    if (lane < 16) { sNv[wv][lane] = nv; sNa[wv][lane] = na; }

    // C[m][0]: rows 0-7 live in lane 0 regs 0-7, rows 8-15 in lane 16.
    // C[m][1]: rows 0-7 in lane 1, rows 8-15 in lane 17.
    if (lane == 0) {
        #pragma unroll
        for (int i = 0; i < 8; ++i) sDotAv[wv][i] = cAv[i];
    }
    if (lane == 16) {
        #pragma unroll
        for (int i = 0; i < 8; ++i) sDotAv[wv][8 + i] = cAv[i];
    }
    if (lane == 1) {
        #pragma unroll
        for (int i = 0; i < 8; ++i) sDotVa[wv][i] = cAa[i];
    }
    if (lane == 17) {
        #pragma unroll
        for (int i = 0; i < 8; ++i) sDotVa[wv][8 + i] = cAa[i];
    }
    __syncthreads();

    const int idx = wsIdx[b];
    if (tid < ROWS_PB) {
        int w2 = tid >> 4, r = tid & 15;
        int t  = chunk * ROWS_PB + tid;
        float dAv   = sDotAv[w2][r];
        float dVa   = sDotVa[w2][r];
        float nrm_v = fmaxf(sqrtf(sNv[w2][r]), EPS_);
        float nrm_a = fmaxf(sqrtf(sNa[w2][r]), EPS_);
        float simA  = dAv / (nrm_v * TEMP_);   // cos(v_t, a_anchor)/T
        float simV  = dVa / (nrm_a * TEMP_);   // cos(a_t, v_anchor)/T
        if (t == idx) wsPos[b] = simA;         // exactly one writer per b
        sEa[tid] = (t == idx) ? 0.0f : expf(simA);
        sEv[tid] = (t == idx) ? 0.0f : expf(simV);
    }
    __syncthreads();
    for (int s = 64; s > 0; s >>= 1) {
        if (tid < s) { sEa[tid] += sEa[tid + s]; sEv[tid] += sEv[tid + s]; }
        __syncthreads();
    }
    if (tid == 0) {
        wsNegA[b * NCHUNK + chunk] = sEa[0];
        wsNegV[b * NCHUNK + chunk] = sEv[0];
    }
}

// ---------------------------------------------------------------------------
// Kernel 3: deterministic finalize. 1 block, 64 threads (one per batch).
// loss_b = 0.5*(log(negA)+log(negV)) - pos ; out = sum_b / B
// ---------------------------------------------------------------------------
__global__ void cl_finalize_kernel(const float* __restrict__ wsPos,
                                   const float* __restrict__ wsNegA,
                                   const float* __restrict__ wsNegV,
                                   float* __restrict__ out)
{
    __shared__ float sRed[64];
    const int tid = threadIdx.x;   // = b
    float na = 0.0f, nv = 0.0f;
    #pragma unroll
    for (int c = 0; c < NCHUNK; ++c) {
        na += wsNegA[tid * NCHUNK + c];
        nv += wsNegV[tid * NCHUNK + c];
    }
    float pos = wsPos[tid];
    sRed[tid] = 0.5f * (logf(na) + logf(nv)) - pos;
    __syncthreads();
    for (int s = 32; s > 0; s >>= 1) {
        if (tid < s) sRed[tid] += sRed[tid + s];
        __syncthreads();
    }
    if (tid == 0) out[0] = sRed[0] / (float)NB;
}

// ---------------------------------------------------------------------------
extern "C" void kernel_launch(void* const* d_in, const int* in_sizes, int n_in,
                              void* d_out, int out_size, void* d_ws, size_t ws_size,
                              hipStream_t stream)
{
    const float* video = (const float*)d_in[0];   // [64,512,512] f32
    const float* audio = (const float*)d_in[1];   // [64,512,512] f32
    const int*   mask  = (const int*)d_in[2];     // [64,512] i32
    float* out = (float*)d_out;

    // workspace layout (floats): idx(64 ints) | anchors(2*64*512) | pos(64)
    //                            | negA(64*4) | negV(64*4)
    int*   wsIdx  = (int*)d_ws;
    float* wsAnch = (float*)d_ws + 64;
    float* wsPos  = wsAnch + 2 * NB * ND;
    float* wsNegA = wsPos + NB;
    float* wsNegV = wsNegA + NB * NCHUNK;

    cl_anchors_kernel<<<dim3(NB), dim3(256), 0, stream>>>(
        video, audio, mask, wsIdx, wsAnch);

    cl_sims_kernel<<<dim3(NCHUNK, NB), dim3(256), 0, stream>>>(
        video, audio, wsIdx, wsAnch, wsPos, wsNegA, wsNegV);

    cl_finalize_kernel<<<dim3(1), dim3(64), 0, stream>>>(
        wsPos, wsNegA, wsNegV, out);
}